// LogOddsPerformanceTransformer_3805341024763
// MI455X (gfx1250) — compile-verified
//
#include <hip/hip_runtime.h>
#include <cstdint>

// LogOdds + bucketize (straight-through estimator value path).
//   out[i] = s - (s - bins[j]),  s = log(x) - log(1-x),
//   j = largest index with bins[j] <= max(s, bins[0]).
//
// Memory-bound: 33.6 MB of HBM traffic -> ~1.4 us floor at 23.3 TB/s.
// Strategy: float4 (b128) streaming, v_log_f32 transcendentals, bins staged
// into LDS via gfx1250 async global->LDS copy, branchless 6-step LDS binary
// search (conflict-free: 64 edges == 64 banks).

#define NBINS   64
#define THREADS 256
#define VPT     4      // float4 chunks per thread (sets the grid size)

#if defined(__gfx1250__) && \
    __has_builtin(__builtin_amdgcn_global_load_async_to_lds_b32) && \
    __has_builtin(__builtin_amdgcn_s_wait_asynccnt)
#define USE_ASYNC_LDS 1
#else
#define USE_ASYNC_LDS 0
#endif

__device__ __forceinline__ float fast_log2(float x) {
#if __has_builtin(__builtin_amdgcn_logf)
  return __builtin_amdgcn_logf(x);   // v_log_f32
#else
  return __log2f(x);
#endif
}

// Returns s - (s - bins[j]) with j = largest index s.t. bins[j] <= max(s, bins[0]).
__device__ __forceinline__ float bucket_st(float x, const float* sbins) {
  const float LN2 = 0.69314718055994530942f;
  // log(x) - log(1-x) = (log2(x) - log2(1-x)) * ln2  -> 2x v_log_f32 + sub + mul
  float s  = (fast_log2(x) - fast_log2(1.0f - x)) * LN2;
  float sb = fmaxf(s, sbins[0]);
  // Branchless binary search: probes 0+{32,16,8,4,2,1} never exceed 63.
  int j = 0;
#pragma unroll
  for (int step = NBINS >> 1; step >= 1; step >>= 1) {
    int c = j + step;
    if (sbins[c] <= sb) j = c;   // v_cmp + v_cndmask after ds_load_b32
  }
  float b = sbins[j];
  // Replicate the reference's exact FP sequence: scores - (scores - scores_bins).
  return s - (s - b);
}

__global__ void __launch_bounds__(THREADS)
logodds_bucketize_kernel(const float4* __restrict__ X4,
                         const float*  __restrict__ bins,
                         float4*       __restrict__ O4,
                         int n4,
                         const float*  __restrict__ Xs,
                         float*        __restrict__ Out,
                         int n) {
  __shared__ float sbins[NBINS];
  const int tid = threadIdx.x;

  // Stage the 64 bin edges (256 B) into LDS.
#if USE_ASYNC_LDS
  if (tid < NBINS) {
    // generic -> AS1 / AS3 via integer casts; the builtin wants (AS1 int*,
    // AS3 int*). AS3 offset = low 32 bits of the generic LDS address.
    auto g = (__attribute__((address_space(1))) int*)(uintptr_t)(bins + tid);
    auto l = (__attribute__((address_space(3))) int*)(uint32_t)(uintptr_t)(&sbins[tid]);
    __builtin_amdgcn_global_load_async_to_lds_b32(g, l, 0, 0);
  }
  __builtin_amdgcn_s_wait_asynccnt(0);
  __syncthreads();
#else
  if (tid < NBINS) sbins[tid] = bins[tid];
  __syncthreads();
#endif

  const int stride = (int)(gridDim.x * blockDim.x);
  for (int i = (int)(blockIdx.x * blockDim.x) + tid; i < n4; i += stride) {
    // Speculative prefetch of the next grid-stride chunk (global_prefetch_b8);
    // OOB prefetches are silently dropped.
    __builtin_prefetch((const void*)(X4 + i + stride), 0, 0);

    float4 x = X4[i];              // global_load_b128
    float4 r;
    r.x = bucket_st(x.x, sbins);
    r.y = bucket_st(x.y, sbins);
    r.z = bucket_st(x.z, sbins);
    r.w = bucket_st(x.w, sbins);
    O4[i] = r;                     // global_store_b128
  }

  // Scalar tail for n not divisible by 4 (not hit for N = 4M, kept for safety).
  const int base = n4 << 2;
  const int ti = base + (int)(blockIdx.x * blockDim.x) + tid;
  if (ti < n) Out[ti] = bucket_st(Xs[ti], sbins);
}

extern "C" void kernel_launch(void* const* d_in, const int* in_sizes, int n_in,
                              void* d_out, int out_size, void* d_ws, size_t ws_size,
                              hipStream_t stream) {
  const float* Xs   = (const float*)d_in[0];
  const float* bins = (const float*)d_in[1];
  float*       out  = (float*)d_out;

  const int n  = in_sizes[0];
  const int n4 = n >> 2;

  int blocks = (n4 + THREADS * VPT - 1) / (THREADS * VPT);
  if (blocks < 1) blocks = 1;

  logodds_bucketize_kernel<<<blocks, THREADS, 0, stream>>>(
      (const float4*)Xs, bins, (float4*)out, n4, Xs, out, n);
}